// MpcCombiner_15522011807803
// MI455X (gfx1250) — compile-verified
//
#include <hip/hip_runtime.h>

// Problem constants from the reference: N=8, B=4096, D=1024, all float32.
#define N_COMP 8
#define BATCH  4096
#define DIM    1024
#define LOG2PI 1.8378770664093454f

typedef __attribute__((ext_vector_type(4))) float fx4;
typedef __attribute__((ext_vector_type(2))) float v2f;
typedef __attribute__((ext_vector_type(8))) float v8f;

// One block per batch row b. 256 threads (8 wave32s); thread t owns the 4
// consecutive D-elements [4t, 4t+4). Streaming, memory-bound: NT b128 loads,
// scan state in registers, NT b128 stores, then a WMMA-based f32 reduction of
// the 256 per-thread log_z partials.
__global__ __launch_bounds__(256)
void MpcCombiner_15522011807803_kernel(const float* __restrict__ means,
                                       const float* __restrict__ logsig,
                                       float* __restrict__ out) {
  const int b   = blockIdx.x;
  const int tid = threadIdx.x;
  const size_t comp = (size_t)BATCH * DIM;               // component stride
  const size_t off  = (size_t)b * DIM + (size_t)tid * 4; // this thread's slice

  __shared__ float s_part[256];

  // ---- component 0: initialize carry (mean, variance) ----
  fx4 mm = __builtin_nontemporal_load((const fx4*)(means  + off));
  fx4 ll = __builtin_nontemporal_load((const fx4*)(logsig + off));
  float m1[4], v1[4], acc[4];
#pragma unroll
  for (int i = 0; i < 4; ++i) {
    m1[i]  = mm[i];
    v1[i]  = __expf(ll[i]);
    acc[i] = 0.0f;
  }

  // ---- sequential combine over components 1..7 ----
#pragma unroll
  for (int n = 1; n < N_COMP; ++n) {
    const float* pm = means  + (size_t)n * comp + off;
    const float* pl = logsig + (size_t)n * comp + off;
    if (n + 1 < N_COMP) {                       // prefetch next 16MB-strided stream
      __builtin_prefetch(pm + comp, 0, 3);      // -> global_prefetch_b8
      __builtin_prefetch(pl + comp, 0, 3);
    }
    fx4 m2v = __builtin_nontemporal_load((const fx4*)pm);
    fx4 l2v = __builtin_nontemporal_load((const fx4*)pl);
#pragma unroll
    for (int i = 0; i < 4; ++i) {
      float m2  = m2v[i];
      float v2  = __expf(l2v[i]);
      float vs  = v1[i] + v2;
      float inv = 1.0f / vs;                     // shared rcp for cmean & cvar
      float cmean = (m1[i] * v2 + m2 * v1[i]) * inv;
      float cvar  = (v1[i] * v2) * inv;
      float s  = vs + 1e-6f;
      float dm = m1[i] - m2;
      acc[i] += dm * dm / s + __logf(s) + LOG2PI;
      m1[i] = cmean;
      v1[i] = cvar;                              // carry variance, not log-var
    }
  }

  // ---- elementwise outputs: post_mean, post_logvar ----
  fx4 om, ov;
#pragma unroll
  for (int i = 0; i < 4; ++i) { om[i] = m1[i]; ov[i] = __logf(v1[i]); }
  __builtin_nontemporal_store(om, (fx4*)(out + off));
  __builtin_nontemporal_store(ov, (fx4*)(out + comp + off));

  // ---- log_z reduction: 256 partials -> 1, via f32 WMMA row-sums ----
  s_part[tid] = acc[0] + acc[1] + acc[2] + acc[3];
  __syncthreads();

  if (tid < 32) {                                // wave 0 only, EXEC all-ones
    const int row = tid & 15;
    const int kc  = (tid >> 4) * 2;              // lanes 0-15: K=0,1; 16-31: K=2,3
    v2f ones; ones[0] = 1.0f; ones[1] = 1.0f;    // B = ones(4x16)
    v8f c = {};
#pragma unroll
    for (int j = 0; j < 4; ++j) {                // 4 accumulating 16x4 A-tiles
      v2f a;
      a[0] = s_part[j * 64 + row * 4 + kc + 0];
      a[1] = s_part[j * 64 + row * 4 + kc + 1];
      // D[m,n] = sum_k A[m,k] + C[m,n]  (f32 in, f32 accumulate)
      c = __builtin_amdgcn_wmma_f32_16x16x4_f32(false, a, false, ones,
                                                (short)0, c, false, false);
    }
    // per-lane: sum C over its 8 M-rows (column N = tid&15, all columns equal)
    float csum = c[0] + c[1] + c[2] + c[3] + c[4] + c[5] + c[6] + c[7];
    s_part[tid] = csum;                          // lane l holds M=0..7 (l<16) / M=8..15
  }
  __syncthreads();
  if (tid == 0) {
    out[2 * comp + (size_t)b] = -0.5f * (s_part[0] + s_part[16]);
  }
}

extern "C" void kernel_launch(void* const* d_in, const int* in_sizes, int n_in,
                              void* d_out, int out_size, void* d_ws, size_t ws_size,
                              hipStream_t stream) {
  (void)in_sizes; (void)n_in; (void)d_ws; (void)ws_size; (void)out_size;
  const float* means  = (const float*)d_in[0];
  const float* logsig = (const float*)d_in[1];
  float* out = (float*)d_out;
  MpcCombiner_15522011807803_kernel<<<BATCH, 256, 0, stream>>>(means, logsig, out);
}